// Attention_111669149864
// MI455X (gfx1250) — compile-verified
//
#include <hip/hip_runtime.h>
#include <math.h>
#include <stdint.h>

#define TT 2048
#define HH 8

typedef __attribute__((ext_vector_type(2))) float v2f;
typedef __attribute__((ext_vector_type(8))) float v8f;

__device__ __forceinline__ v8f wmma4(v2f a, v2f b, v8f c) {
  // V_WMMA_F32_16X16X4_F32 : D = A(16x4 f32) x B(4x16 f32) + C(16x16 f32)
  return __builtin_amdgcn_wmma_f32_16x16x4_f32(false, a, false, b, (short)0, c, false, false);
}

// CDNA5 async copy: global -> LDS, 16B per lane, tracked by ASYNCcnt.
// GVS mode: uniform 64-bit SGPR base + per-lane 32-bit byte offset; LDS byte
// address supplied in a VGPR (hardware adds LDS_BASE).
__device__ __forceinline__ void async_ld_b128(uint32_t ldsOff, uint32_t gOff,
                                              const float* base) {
  asm volatile("global_load_async_to_lds_b128 %0, %1, %2"
               :: "v"(ldsOff), "v"(gOff), "s"(base)
               : "memory");
}
__device__ __forceinline__ void wait_async0() {
  asm volatile("s_wait_asynccnt 0" ::: "memory");
}
__device__ __forceinline__ uint32_t lds_off(const void* p) {
  return (uint32_t)(uintptr_t)p;
}

// ---------------------------------------------------------------------------
// Generic one-wave GEMM: D[16,N-tile] = act(alpha * A[16,K] x B[K,N] + bias)
// grid.x = N/16 tiles, grid.y = batched slices (strides in elements).
// BT: B stored [N][K] (transposed) ; ACT: 1 = leaky_relu(0.2)
// ---------------------------------------------------------------------------
template<int BT, int ACT>
__global__ __launch_bounds__(32) void gemm16_kernel(
    const float* __restrict__ A, int lda, long long aStrideY,
    const float* __restrict__ B, int ldb, long long bStrideY,
    const float* __restrict__ bias, long long biasStrideY,
    float* __restrict__ D, int ldd, long long dStrideY,
    int K, float alpha) {
  int n0 = blockIdx.x * 16;
  int y  = blockIdx.y;
  const float* A_ = A + (size_t)y * aStrideY;
  const float* B_ = B + (size_t)y * bStrideY;
  float*       D_ = D + (size_t)y * dStrideY;
  int lane = threadIdx.x;
  int m = lane & 15;              // A row / B col within tile
  int kb2 = (lane >> 4) << 1;     // K-pair selector per half-wave
  int n = m;
  v8f acc = {};
#pragma unroll 4
  for (int k = 0; k < K; k += 4) {
    float2 af = *(const float2*)(A_ + (size_t)m * lda + k + kb2);
    v2f a; a.x = af.x; a.y = af.y;
    v2f bv;
    if (BT) {
      float2 bf = *(const float2*)(B_ + (size_t)(n0 + n) * ldb + k + kb2);
      bv.x = bf.x; bv.y = bf.y;
    } else {
      bv.x = B_[(size_t)(k + kb2) * ldb + n0 + n];
      bv.y = B_[(size_t)(k + kb2 + 1) * ldb + n0 + n];
    }
    acc = wmma4(a, bv, acc);
  }
  const float* bias_ = bias ? bias + (size_t)y * biasStrideY : nullptr;
#pragma unroll
  for (int r = 0; r < 8; ++r) {
    int mo = r + ((lane >> 4) << 3);        // D row = r + 8*(lane>=16)
    float v = acc[r] * alpha;
    if (bias_) v += bias_[n0 + n];
    if (ACT == 1) v = v > 0.f ? v : 0.2f * v;   // leaky_relu(0.2)
    D_[(size_t)mo * ldd + n0 + n] = v;
  }
}

// ---------------------------------------------------------------------------
// scores_raw[b][t][h(16-pad)] = keys[b,t,:] . c[b,h,:]   (c pre-scaled 1/sqrt(dk))
// Block: b = blockIdx.y, 128 t-rows (8 waves x 16). keys double-buffered in
// LDS via async loads overlapping WMMA compute. cL padded to 16 rows with
// zeros so the B-read needs no exec-mask guard in the hot loop.
// ---------------------------------------------------------------------------
__global__ __launch_bounds__(256) void scores_kernel(
    const float* __restrict__ keys, const float* __restrict__ cbuf,
    float* __restrict__ spad) {
  __shared__ float cL[16 * 1032];       // c[b] as [h][j] pad stride 1032, rows 8..15 = 0
  __shared__ float kT[2][128 * 34];     // keys tiles [t][k] pad stride 34
  int b = blockIdx.y;
  int tBase = blockIdx.x * 128;
  int tid = threadIdx.x;
  const float* cb = cbuf + (size_t)b * 8192;

  // async-load c[b] rows 0..7 (8192 floats, 16B per lane per round)
  for (int i = tid; i < 2048; i += 256) {
    int h = i >> 8, j4 = (i & 255) << 2;
    async_ld_b128(lds_off(&cL[h * 1032 + j4]), (uint32_t)((h * 1024 + j4) * 4), cb);
  }
  // zero pad rows 8..15 (plain LDS stores, covered by the barrier)
  for (int i = tid; i < 2048; i += 256) {
    int h = 8 + (i >> 8), j4 = (i & 255) << 2;
    float* dst = &cL[h * 1032 + j4];
    dst[0] = 0.f; dst[1] = 0.f; dst[2] = 0.f; dst[3] = 0.f;
  }

  int lane = tid & 31, wid = tid >> 5;
  int m = lane & 15, kb2 = (lane >> 4) << 1, n = m;
  const float* kp = keys + (size_t)b * TT * 1024;
  int r_ = tid >> 3, c4_ = (tid & 7) << 2;       // 128x32 tile slots

  // prologue: async-load chunk 0 into buffer 0
#pragma unroll
  for (int i = 0; i < 4; ++i) {
    int r = r_ + i * 32;
    async_ld_b128(lds_off(&kT[0][r * 34 + c4_]),
                  (uint32_t)(((tBase + r) * 1024 + c4_) * 4), kp);
  }
  wait_async0();
  __syncthreads();

  v8f acc = {};
  for (int k0 = 0; k0 < 1024; k0 += 32) {
    int cur = (k0 >> 5) & 1;
    if (k0 + 32 < 1024) {                        // prefetch next chunk
#pragma unroll
      for (int i = 0; i < 4; ++i) {
        int r = r_ + i * 32;
        async_ld_b128(lds_off(&kT[cur ^ 1][r * 34 + c4_]),
                      (uint32_t)(((tBase + r) * 1024 + k0 + 32 + c4_) * 4), kp);
      }
    }
    int rowBase = (wid * 16 + m) * 34;
    const float* cRow = &cL[n * 1032 + k0 + kb2];
#pragma unroll
    for (int kk = 0; kk < 32; kk += 4) {
      float2 av = *(const float2*)&kT[cur][rowBase + kk + kb2];
      float2 bf = *(const float2*)(cRow + kk);
      v2f a; a.x = av.x; a.y = av.y;
      v2f bv; bv.x = bf.x; bv.y = bf.y;
      acc = wmma4(a, bv, acc);
    }
    wait_async0();
    __syncthreads();
  }
#pragma unroll
  for (int r = 0; r < 8; ++r) {
    int mo = r + ((lane >> 4) << 3);
    int t = tBase + wid * 16 + mo;
    spad[((size_t)(b * TT + t)) * 16 + n] = acc[r];
  }
}

// ---------------------------------------------------------------------------
// Softmax over t per (b,h), in-place on padded score buffer. Handles the
// start/end mask and the fully-masked -> uniform special case.
// ---------------------------------------------------------------------------
__global__ __launch_bounds__(256) void softmax_kernel(
    float* __restrict__ spad, const float* __restrict__ start,
    const float* __restrict__ endi) {
  __shared__ float red[256];
  int bh = blockIdx.x, b = bh >> 3, h = bh & 7, tid = threadIdx.x;
  float lo = floorf(start[b]), hi = ceilf(endi[b]);
  bool allMasked = (lo > hi) || (hi < 0.f) || (lo > (float)(TT - 1));
  float v[8];
  float mx = -3.4e38f;
  for (int i = 0; i < 8; ++i) {
    int t = tid + i * 256;
    float raw = spad[((size_t)(b * TT + t)) * 16 + h];
    float ft = (float)t;
    bool msk = (ft < lo) || (ft > hi);
    float val = allMasked ? 1.0f : (msk ? -INFINITY : raw);
    v[i] = val; mx = fmaxf(mx, val);
  }
  red[tid] = mx; __syncthreads();
  for (int s = 128; s > 0; s >>= 1) { if (tid < s) red[tid] = fmaxf(red[tid], red[tid + s]); __syncthreads(); }
  mx = red[0]; __syncthreads();
  float sum = 0.f;
  for (int i = 0; i < 8; ++i) { float e = expf(v[i] - mx); v[i] = e; sum += e; }
  red[tid] = sum; __syncthreads();
  for (int s = 128; s > 0; s >>= 1) { if (tid < s) red[tid] += red[tid + s]; __syncthreads(); }
  float inv = 1.f / red[0];
  for (int i = 0; i < 8; ++i) {
    int t = tid + i * 256;
    spad[((size_t)(b * TT + t)) * 16 + h] = v[i] * inv;
  }
}

// ---------------------------------------------------------------------------
// u[b][h][j] = sum_t s[b,t,h] * values[b,t,j]  (A = padded scores, M=8 of 16)
// values double-buffered in LDS via async loads; A-chunk preloaded to regs.
// ---------------------------------------------------------------------------
__global__ __launch_bounds__(256) void u_kernel(
    const float* __restrict__ values, const float* __restrict__ spad,
    float* __restrict__ ubuf) {
  __shared__ float vT[2][32 * 140];     // values tiles [k][j] pad stride 140
  int b = blockIdx.y;
  int j0blk = blockIdx.x * 128;
  int tid = threadIdx.x, lane = tid & 31, wid = tid >> 5;
  int m = lane & 15, kb2 = (lane >> 4) << 1, n = m;
  const float* vp = values + (size_t)b * TT * 1024;
  const float* sp = spad + (size_t)b * TT * 16;
  int r_ = tid >> 5, c4_ = (tid & 31) << 2;      // 32x128 tile slots

#pragma unroll
  for (int i = 0; i < 4; ++i) {
    int r = r_ + i * 8;
    async_ld_b128(lds_off(&vT[0][r * 140 + c4_]),
                  (uint32_t)((r * 1024 + j0blk + c4_) * 4), vp);
  }
  wait_async0();
  __syncthreads();

  v8f acc = {};
  for (int k0 = 0; k0 < TT; k0 += 32) {
    int cur = (k0 >> 5) & 1;
    // preload this chunk's A values into registers (one clustered load burst)
    float2 aReg[8];
#pragma unroll
    for (int q = 0; q < 8; ++q) {
      int kr = k0 + q * 4 + kb2;
      aReg[q].x = sp[(size_t)kr * 16 + m];       // pads (m>=8) are exact zeros
      aReg[q].y = sp[(size_t)(kr + 1) * 16 + m];
    }
    if (k0 + 32 < TT) {                          // prefetch next values chunk
#pragma unroll
      for (int i = 0; i < 4; ++i) {
        int r = r_ + i * 8;
        async_ld_b128(lds_off(&vT[cur ^ 1][r * 140 + c4_]),
                      (uint32_t)(((k0 + 32 + r) * 1024 + j0blk + c4_) * 4), vp);
      }
    }
#pragma unroll
    for (int q = 0; q < 8; ++q) {
      int kk = q * 4;
      v2f a; a.x = aReg[q].x; a.y = aReg[q].y;
      v2f bv;
      bv.x = vT[cur][(kk + kb2) * 140 + wid * 16 + n];
      bv.y = vT[cur][(kk + kb2 + 1) * 140 + wid * 16 + n];
      acc = wmma4(a, bv, acc);
    }
    wait_async0();
    __syncthreads();
  }
#pragma unroll
  for (int r = 0; r < 8; ++r) {
    int mo = r + ((lane >> 4) << 3);
    if (mo < 8)
      ubuf[(size_t)(b * 8 + mo) * 1024 + j0blk + wid * 16 + n] = acc[r];
  }
}

// ---------------------------------------------------------------------------
// LayerNorm over last dim (1024), optional fused residual add. One block/row.
// ---------------------------------------------------------------------------
__global__ __launch_bounds__(256) void ln_kernel(
    const float* __restrict__ X, const float* __restrict__ Add,
    float* __restrict__ Out) {
  __shared__ float red[256];
  int b = blockIdx.x, tid = threadIdx.x;
  float v[4];
  float s = 0.f;
  for (int i = 0; i < 4; ++i) {
    int j = tid + i * 256;
    float x = X[b * 1024 + j];
    if (Add) x += Add[b * 1024 + j];
    v[i] = x; s += x;
  }
  red[tid] = s; __syncthreads();
  for (int st = 128; st > 0; st >>= 1) { if (tid < st) red[tid] += red[tid + st]; __syncthreads(); }
  float mean = red[0] * (1.f / 1024.f);
  __syncthreads();
  float s2 = 0.f;
  for (int i = 0; i < 4; ++i) { float d = v[i] - mean; s2 += d * d; }
  red[tid] = s2; __syncthreads();
  for (int st = 128; st > 0; st >>= 1) { if (tid < st) red[tid] += red[tid + st]; __syncthreads(); }
  float inv = 1.f / sqrtf(red[0] * (1.f / 1024.f) + 1e-5f);
  for (int i = 0; i < 4; ++i) {
    int j = tid + i * 256;
    Out[b * 1024 + j] = (v[i] - mean) * inv;
  }
}

// attw[b][t] = mean over h of softmaxed scores
__global__ __launch_bounds__(256) void attw_kernel(
    const float* __restrict__ spad, float* __restrict__ outw) {
  int idx = blockIdx.x * 256 + threadIdx.x;   // over B*T
  const float* p = spad + (size_t)idx * 16;
  float s = 0.f;
  for (int h = 0; h < 8; ++h) s += p[h];
  outw[idx] = s * 0.125f;
}

// ---------------------------------------------------------------------------
extern "C" void kernel_launch(void* const* d_in, const int* in_sizes, int n_in,
                              void* d_out, int out_size, void* d_ws, size_t ws_size,
                              hipStream_t stream) {
  (void)out_size; (void)ws_size;
  const float* values = (const float*)d_in[0];
  const float* keys   = (const float*)d_in[1];
  const float* qinput = (const float*)d_in[2];
  const float* start  = (const float*)d_in[3];
  const float* endi   = (const float*)d_in[4];

  // Detect params flattening order via the unique 2048x1024 query_net W0.
  int qnetW0 = -1;
  for (int i = 5; i < n_in; ++i) if (in_sizes[i] == 2048 * 1024) { qnetW0 = i; break; }

  int qnW[5], qnB[5], lq[2], lqb[2], lk[2], lkb[2], lv[2], lvb[2], lao[2], laob[2];
  int lpW[2][4], lpB[2][4], oW, oB;
  if (qnetW0 != 39) {
    // dict insertion-order flattening: query_net, layers[{q,k,v,attn_out,pred}], out
    for (int i = 0; i < 5; ++i) { qnW[i] = 5 + 2 * i; qnB[i] = 6 + 2 * i; }
    for (int l = 0; l < 2; ++l) {
      int base = 15 + 16 * l;
      lq[l] = base + 0; lqb[l] = base + 1; lk[l] = base + 2; lkb[l] = base + 3;
      lv[l] = base + 4; lvb[l] = base + 5; lao[l] = base + 6; laob[l] = base + 7;
      for (int i = 0; i < 4; ++i) { lpW[l][i] = base + 8 + 2 * i; lpB[l][i] = base + 9 + 2 * i; }
    }
    oW = 47; oB = 48;
  } else {
    // jax alphabetical flattening: layers[{attn_out,k,pred,q,v}], out, query_net
    for (int l = 0; l < 2; ++l) {
      int base = 5 + 16 * l;
      lao[l] = base + 0; laob[l] = base + 1; lk[l] = base + 2; lkb[l] = base + 3;
      for (int i = 0; i < 4; ++i) { lpW[l][i] = base + 4 + 2 * i; lpB[l][i] = base + 5 + 2 * i; }
      lq[l] = base + 12; lqb[l] = base + 13; lv[l] = base + 14; lvb[l] = base + 15;
    }
    oW = 37; oB = 38;
    for (int i = 0; i < 5; ++i) { qnW[i] = 39 + 2 * i; qnB[i] = 40 + 2 * i; }
  }
  (void)lkb; (void)lvb;  // k bias is softmax-invariant; v bias folded below

  float* ws = (float*)d_ws;
  float* t0    = ws;                     // [16*2048]
  float* t1    = t0 + 16 * 2048;        // [16*2048]
  float* query = t1 + 16 * 2048;        // [16*1024]
  float* qlin  = query + 16 * 1024;     // [16*1024]
  float* cbuf  = qlin + 16 * 1024;      // [16*8*1024]
  float* spad  = cbuf + 16 * 8 * 1024;  // [16*2048*16]
  float* ubuf  = spad + 16 * 2048 * 16; // [16*8*1024]
  float* vals  = ubuf + 16 * 8 * 1024;  // [16*1024]
  float* rawb  = vals + 16 * 1024;      // [16*1024]
  float* xln   = rawb + 16 * 1024;      // [16*1024]

  float* outp = (float*)d_out;          // [16*1024]
  float* attw = outp + 16 * 1024;       // [16*2048]

#define FP(i) ((const float*)d_in[i])
  // ---- query_net MLP: 2048->1024, 3x 1024->1024 (leaky 0.2), final plain
  gemm16_kernel<0,1><<<dim3(64, 1), 32, 0, stream>>>(qinput, 2048, 0, FP(qnW[0]), 1024, 0, FP(qnB[0]), 0, t0, 1024, 0, 2048, 1.f);
  gemm16_kernel<0,1><<<dim3(64, 1), 32, 0, stream>>>(t0, 1024, 0, FP(qnW[1]), 1024, 0, FP(qnB[1]), 0, t1, 1024, 0, 1024, 1.f);
  gemm16_kernel<0,1><<<dim3(64, 1), 32, 0, stream>>>(t1, 1024, 0, FP(qnW[2]), 1024, 0, FP(qnB[2]), 0, t0, 1024, 0, 1024, 1.f);
  gemm16_kernel<0,1><<<dim3(64, 1), 32, 0, stream>>>(t0, 1024, 0, FP(qnW[3]), 1024, 0, FP(qnB[3]), 0, t1, 1024, 0, 1024, 1.f);
  gemm16_kernel<0,0><<<dim3(64, 1), 32, 0, stream>>>(t1, 1024, 0, FP(qnW[4]), 1024, 0, FP(qnB[4]), 0, query, 1024, 0, 1024, 1.f);

  const float invsq = 0.08838834764831845f;   // 1/sqrt(dk=128), TEMP=1
  for (int l = 0; l < 2; ++l) {
    // q = query @ Wq + bq
    gemm16_kernel<0,0><<<dim3(64, 1), 32, 0, stream>>>(query, 1024, 0, FP(lq[l]), 1024, 0, FP(lqb[l]), 0, qlin, 1024, 0, 1024, 1.f);
    // c[b][h][j] = invsq * sum_d Wk[j, h*128+d] * q[b, h*128+d]  (Bt path, grid.y = h)
    gemm16_kernel<1,0><<<dim3(64, 8), 32, 0, stream>>>(qlin, 1024, 128, FP(lk[l]), 1024, 128, nullptr, 0, cbuf, 8192, 1024, 128, invsq);
    // scores / softmax / u
    scores_kernel<<<dim3(16, 16), 256, 0, stream>>>(keys, cbuf, spad);
    softmax_kernel<<<dim3(128), 256, 0, stream>>>(spad, start, endi);
    u_kernel<<<dim3(8, 16), 256, 0, stream>>>(values, spad, ubuf);
    // vals[b, h*128+d] = u[b,h,:] @ Wv[:, h*128+d] + bv   (grid.y = h)
    gemm16_kernel<0,0><<<dim3(8, 8), 32, 0, stream>>>(ubuf, 8192, 1024, FP(lv[l]), 1024, 128, FP(lvb[l]), 128, vals, 1024, 128, 1024, 1.f);
    // raw = attn_out(vals)
    gemm16_kernel<0,0><<<dim3(64, 1), 32, 0, stream>>>(vals, 1024, 0, FP(lao[l]), 1024, 0, FP(laob[l]), 0, rawb, 1024, 0, 1024, 1.f);
    ln_kernel<<<dim3(16), 256, 0, stream>>>(rawb, nullptr, xln);
    // predictor MLP (3 leaky + plain), then query = ln(pred + query)
    gemm16_kernel<0,1><<<dim3(64, 1), 32, 0, stream>>>(xln, 1024, 0, FP(lpW[l][0]), 1024, 0, FP(lpB[l][0]), 0, t0, 1024, 0, 1024, 1.f);
    gemm16_kernel<0,1><<<dim3(64, 1), 32, 0, stream>>>(t0, 1024, 0, FP(lpW[l][1]), 1024, 0, FP(lpB[l][1]), 0, t1, 1024, 0, 1024, 1.f);
    gemm16_kernel<0,1><<<dim3(64, 1), 32, 0, stream>>>(t1, 1024, 0, FP(lpW[l][2]), 1024, 0, FP(lpB[l][2]), 0, t0, 1024, 0, 1024, 1.f);
    gemm16_kernel<0,0><<<dim3(64, 1), 32, 0, stream>>>(t0, 1024, 0, FP(lpW[l][3]), 1024, 0, FP(lpB[l][3]), 0, t1, 1024, 0, 1024, 1.f);
    ln_kernel<<<dim3(16), 256, 0, stream>>>(t1, query, query);
  }
  // out = raw @ Wout + bout ; attw = head-mean of last layer's scores
  gemm16_kernel<0,0><<<dim3(64, 1), 32, 0, stream>>>(rawb, 1024, 0, FP(oW), 1024, 0, FP(oB), 0, outp, 1024, 0, 1024, 1.f);
  attw_kernel<<<dim3(128), 256, 0, stream>>>(spad, attw);
#undef FP
}